// RANGE2BEV_45844480918023
// MI455X (gfx1250) — compile-verified
//
#include <hip/hip_runtime.h>
#include <cstdint>
#include <cstddef>

// ---------------- problem constants (fixed by setup_inputs) ----------------
constexpr int   B_   = 4;
constexpr int   C1_  = 32;       // range_res channels
constexpr int   HR_  = 64;
constexpr int   WR_  = 2048;
constexpr int   C2_  = 5;        // scan channels
constexpr int   N_   = 131072;   // points per batch
constexpr int   CT_  = C2_ + C1_;// 37 output channels
constexpr int   W_   = 800;      // (40 - (-40)) / 0.1
constexpr int   H_   = 704;      // 70.4 / 0.1
constexpr int   XOFF = -400;     // floor(-40 / 0.1)
constexpr int   YOFF = 704;      // ceil(70.4 / 0.1)
constexpr float RES_ = 0.1f;

constexpr int    PTS        = 256;                       // points per block
constexpr int    BLKS_PER_B = N_ / PTS;                  // 512
constexpr size_t OUT_ELEMS  = (size_t)B_ * CT_ * W_ * H_;  // 83,353,600 floats
constexpr size_t GRID_CELLS = (size_t)B_ * H_ * W_;        // 2,252,800 cells

static_assert(OUT_ELEMS % 4 == 0, "out must be float4-divisible");
static_assert(GRID_CELLS % 4 == 0, "winner must be uint4-divisible");

typedef float    v4f __attribute__((ext_vector_type(4)));
typedef unsigned v4u __attribute__((ext_vector_type(4)));
typedef unsigned v8u __attribute__((ext_vector_type(8)));

__device__ __forceinline__ int clampi(int v, int lo, int hi) {
    return v < lo ? lo : (v > hi ? hi : v);
}

// BEV cell from point (x, y); matches jnp truncating astype(int32) + clip.
__device__ __forceinline__ void bev_cell(float x, float y, int& xc, int& yc) {
    int xi = (int)((-y) / RES_) - XOFF;
    int yi = (int)((-x) / RES_) + YOFF;
    xc = clampi(xi, 0, W_ - 1);
    yc = clampi(yi, 0, H_ - 1);
}

// ---------------- kernel 1: wide non-temporal clear ----------------
__global__ void RANGE2BEV_zero4(v4f* __restrict__ p, size_t n4) {
    size_t i      = (size_t)blockIdx.x * blockDim.x + threadIdx.x;
    size_t stride = (size_t)gridDim.x * blockDim.x;
    v4f z = {0.0f, 0.0f, 0.0f, 0.0f};
    for (; i < n4; i += stride)
        __builtin_nontemporal_store(z, p + i);
}

// ---------------- kernel 2: winner resolution (deterministic .set) ---------
// winner[cell] = max(n+1) over points landing in cell; 0 == untouched.
// Stages the two needed scan channels via the gfx1250 async global->LDS path.
__global__ void RANGE2BEV_winner(const float* __restrict__ scan,
                                 unsigned* __restrict__ winner) {
    __shared__ float sm[2 * PTS];              // 2 KB: [channel][point]
    const int b  = blockIdx.x / BLKS_PER_B;
    const int p0 = (blockIdx.x % BLKS_PER_B) * PTS;
    const int t  = threadIdx.x;
    const float* sb = scan + (size_t)b * C2_ * N_ + p0;

    // 2 channels x 256 floats = 2048 B = 128 B128 lane-transfers (ASYNCcnt).
    if (t < 2 * PTS / 4) {
        int ch = t >> 6;                       // 64 transfers per channel
        int j  = (t & 63) << 2;                // float index within channel run
        uint64_t gaddr = (uint64_t)(uintptr_t)(sb + (size_t)ch * N_ + j);
        uint32_t laddr = (uint32_t)(uintptr_t)(&sm[ch * PTS + j]);
        asm volatile("global_load_async_to_lds_b128 %0, %1, off"
                     :: "v"(laddr), "v"(gaddr)
                     : "memory");
    }
    asm volatile("s_wait_asynccnt 0" ::: "memory");
    __syncthreads();

    const int n = p0 + t;
    float x = sm[0 * PTS + t];
    float y = sm[1 * PTS + t];
    int xc, yc;
    bev_cell(x, y, xc, yc);
    atomicMax(&winner[((size_t)b * H_ + yc) * W_ + xc], (unsigned)(n + 1));
}

// ---------------- kernel 3: gather + scatter (TDM-staged) ----------------
// One TDM descriptor per block moves the 2D scan tile (5 rows x 256 floats,
// row stride N) into LDS: tensor_load_to_lds + s_wait_tensorcnt (TENSORcnt).
// Winning points then gather 32 range channels (L2-resident strided gather)
// and scatter 37 outputs.
template <bool USE_WINNER>
__global__ void RANGE2BEV_scatter(const float* __restrict__ range_res,
                                  const float* __restrict__ scan,
                                  const int*   __restrict__ range_y,
                                  const int*   __restrict__ range_x,
                                  const unsigned* __restrict__ winner,
                                  float* __restrict__ out) {
    __shared__ float sm[C2_ * PTS];            // 5 KB: [channel][point], contiguous

    const int b  = blockIdx.x / BLKS_PER_B;
    const int p0 = (blockIdx.x % BLKS_PER_B) * PTS;
    const int t  = threadIdx.x;

    const float* sb = scan + (size_t)b * C2_ * N_ + p0;

    if (t < warpSize) {                        // wave 0 issues the TDM op
        const uint64_t gaddr = (uint64_t)(uintptr_t)sb;
        const uint32_t laddr = (uint32_t)(uintptr_t)sm;
        // ---- D# group 0 (4 SGPRs) ----
        v4u g0;
        g0[0] = 1u;                                          // count=1, user mode
        g0[1] = laddr;                                       // lds_addr (bytes)
        g0[2] = (uint32_t)gaddr;                             // global_addr[31:0]
        g0[3] = (uint32_t)((gaddr >> 32) & 0x01FFFFFFu)      // global_addr[56:32]
              | (2u << 30);                                  // type = 2 ("image")
        // ---- D# group 1 (8 SGPRs) ----
        v8u g1;
        g1[0] = 2u << 16;                  // wg_mask=0 | data_size=2 (4 bytes)
        g1[1] = (uint32_t)PTS << 16;       // tensor_dim0[15:0] = 256
        g1[2] = (uint32_t)C2_ << 16;       // tensor_dim0[31:16]=0 | tensor_dim1[15:0]=5
        g1[3] = (uint32_t)PTS << 16;       // tensor_dim1[31:16]=0 | tile_dim0 = 256
        g1[4] = (uint32_t)C2_;             // tile_dim1 = 5 | tile_dim2 = 0
        g1[5] = (uint32_t)N_;              // tensor_dim0_stride[31:0] = 131072
        g1[6] = 0u;                        // stride0[47:32]=0 | dim1_stride[15:0]=0
        g1[7] = 0u;                        // dim1_stride[47:16] = 0
        // 2D tensor: VADDR2/VADDR3 omitted (NULL)
        asm volatile("tensor_load_to_lds %0, %1"
                     :: "s"(g0), "s"(g1)
                     : "memory");
        __builtin_amdgcn_s_wait_tensorcnt(0);  // TDM transfer complete (this wave)
    }
    __syncthreads();                           // LDS tile visible to all waves

    const int n = p0 + t;
    const float x  = sm[0 * PTS + t];
    const float y  = sm[1 * PTS + t];
    const float z  = sm[2 * PTS + t];
    const float c3 = sm[3 * PTS + t];
    const float c4 = sm[4 * PTS + t];

    const int py = range_y[(size_t)b * N_ + n];
    const int px = range_x[(size_t)b * N_ + n];
    const float* rr = range_res + (size_t)b * C1_ * HR_ * WR_
                                + (size_t)py * WR_ + px;
    __builtin_prefetch(rr, 0, 1);              // global_prefetch_b8: warm gather base

    int xc, yc;
    bev_cell(x, y, xc, yc);
    const size_t cell = ((size_t)b * H_ + yc) * W_ + xc;

    bool write = true;
    if (USE_WINNER) write = (winner[cell] == (unsigned)(n + 1));
    if (!write) return;

    const float zc = fminf(fmaxf(z, -3.0f), 1.0f);

    // out[b][ch][xc][yc], dims (B, 37, 800, 704); channel stride = W*H
    float* ob = out + (((size_t)b * CT_) * W_ + xc) * H_ + yc;
    constexpr size_t CS = (size_t)W_ * H_;
    ob[0 * CS] = zc * x;
    ob[1 * CS] = zc * y;
    ob[2 * CS] = zc * z;
    ob[3 * CS] = zc * c3;
    ob[4 * CS] = zc * c4;
#pragma unroll
    for (int c = 0; c < C1_; ++c) {
        float v = rr[(size_t)c * HR_ * WR_];   // strided gather; planes sit in L2
        ob[(size_t)(C2_ + c) * CS] = zc * v;
    }
}

// ---------------- host launch ----------------
extern "C" void kernel_launch(void* const* d_in, const int* in_sizes, int n_in,
                              void* d_out, int out_size, void* d_ws, size_t ws_size,
                              hipStream_t stream) {
    (void)in_sizes; (void)n_in; (void)out_size;
    const float* range_res = (const float*)d_in[0];
    const float* scan      = (const float*)d_in[1];
    const int*   range_y   = (const int*)d_in[2];
    const int*   range_x   = (const int*)d_in[3];
    float*       out       = (float*)d_out;

    // 1) clear 333 MB output (bandwidth floor of this op)
    RANGE2BEV_zero4<<<8192, 256, 0, stream>>>((v4f*)out, OUT_ELEMS / 4);

    const size_t winner_bytes = GRID_CELLS * sizeof(unsigned);  // ~9 MB
    const int    grid = B_ * BLKS_PER_B;                        // 2048 blocks

    if (ws_size >= winner_bytes) {
        unsigned* winner = (unsigned*)d_ws;
        // 2) clear winner grid (re-zeroed every call; harness doesn't re-poison)
        RANGE2BEV_zero4<<<2048, 256, 0, stream>>>((v4f*)winner, GRID_CELLS / 4);
        // 3) deterministic per-cell winner (largest point index wins)
        RANGE2BEV_winner<<<grid, PTS, 0, stream>>>(scan, winner);
        // 4) gather + scatter only winning points
        RANGE2BEV_scatter<true><<<grid, PTS, 0, stream>>>(
            range_res, scan, range_y, range_x, winner, out);
    } else {
        // workspace too small: direct scatter (duplicate cells race, matching
        // XLA's own unordered duplicate-index .set semantics)
        RANGE2BEV_scatter<false><<<grid, PTS, 0, stream>>>(
            range_res, scan, range_y, range_x, nullptr, out);
    }
}